// NetVLADPlusPlus_65575560675841
// MI455X (gfx1250) — compile-verified
//
#include <hip/hip_runtime.h>

typedef __attribute__((ext_vector_type(2))) float v2f;
typedef __attribute__((ext_vector_type(4))) float f4;
typedef __attribute__((ext_vector_type(8))) float v8f;

#define WB 32
#define WT 2048
#define WD 512
#define WK 64
#define NEPS 1e-12f
#define PITCH 68   // 68*4 bytes: 16B aligned rows, stride%64banks==4 -> conflict-free frag reads

// ---------------------------------------------------------------------------
// Pass 1: logits = X * W^T + b, softmax over K, write a[b][t][k]
// grid (T/64, B), block 256 (8 waves); each wave owns 2 16x16 C tiles
// ---------------------------------------------------------------------------
__global__ __launch_bounds__(256) void k_logits_softmax(
    const float* __restrict__ x, const float* __restrict__ conv_w,
    const float* __restrict__ conv_b, float* __restrict__ a_out)
{
    __shared__ float xT[64 * PITCH];   // [t][d]
    __shared__ float wT[64 * PITCH];   // [k][d]
    __shared__ float Lb[64 * PITCH];   // logits [t][k]
    __shared__ float bias[64];

    const int tid  = threadIdx.x;
    const int wave = tid >> 5;
    const int lane = tid & 31;
    const int half = lane >> 4;
    const int lm   = lane & 15;
    const int b    = blockIdx.y;
    const int t0   = blockIdx.x * 64;

    if (tid < 64) bias[tid] = conv_b[tid];

    const int mt  = wave & 3;                 // t-tile of this wave
    const int n0a = ((wave >> 2) * 2) * 16;   // first k-tile col base
    const int n0b = n0a + 16;

    v8f acc0 = {}; v8f acc1 = {};

    for (int d0 = 0; d0 < WD; d0 += 64) {
        // stage X tile (64t x 64d) and W tile (64k x 64d), float4 per thread x4
        for (int i = 0; i < 4; ++i) {
            int idx = tid + i * 256;          // 0..1023
            int row = idx >> 4;
            int c4  = (idx & 15) << 2;
            f4 xv = *(const f4*)(x + ((size_t)b * WT + t0 + row) * WD + d0 + c4);
            *(f4*)(&xT[row * PITCH + c4]) = xv;
            f4 wv = *(const f4*)(conv_w + (size_t)row * WD + d0 + c4);
            *(f4*)(&wT[row * PITCH + c4]) = wv;
        }
        __syncthreads();
        #pragma unroll
        for (int dd = 0; dd < 64; dd += 4) {
            const int kk = dd + 2 * half;     // K index per A/B f32 layout
            v2f af;  af[0]  = xT[(mt * 16 + lm) * PITCH + kk];
                     af[1]  = xT[(mt * 16 + lm) * PITCH + kk + 1];
            v2f bf0; bf0[0] = wT[(n0a + lm) * PITCH + kk];
                     bf0[1] = wT[(n0a + lm) * PITCH + kk + 1];
            acc0 = __builtin_amdgcn_wmma_f32_16x16x4_f32(false, af, false, bf0,
                                                         (short)0, acc0, false, false);
            v2f bf1; bf1[0] = wT[(n0b + lm) * PITCH + kk];
                     bf1[1] = wT[(n0b + lm) * PITCH + kk + 1];
            acc1 = __builtin_amdgcn_wmma_f32_16x16x4_f32(false, af, false, bf1,
                                                         (short)0, acc1, false, false);
        }
        __syncthreads();
    }

    // spill C tiles: c[i] -> (t = mt*16 + i + 8*half, k = ncol + lm)
    #pragma unroll
    for (int i = 0; i < 8; ++i) {
        int r = mt * 16 + i + 8 * half;
        Lb[r * PITCH + n0a + lm] = acc0[i];
        Lb[r * PITCH + n0b + lm] = acc1[i];
    }
    __syncthreads();

    // softmax over K per row t (threads 0..63)
    if (tid < 64) {
        const int t = tid;
        float m = -3.0e38f;
        #pragma unroll 8
        for (int k = 0; k < 64; ++k)
            m = fmaxf(m, Lb[t * PITCH + k] + bias[k]);
        float s = 0.f;
        #pragma unroll 8
        for (int k = 0; k < 64; ++k) {
            float e = expf(Lb[t * PITCH + k] + bias[k] - m);
            Lb[t * PITCH + k] = e;
            s += e;
        }
        float inv = 1.0f / s;
        float* dst = a_out + ((size_t)b * WT + t0 + t) * WK;
        #pragma unroll 8
        for (int k = 0; k < 64; ++k) dst[k] = Lb[t * PITCH + k] * inv;
    }
}

// ---------------------------------------------------------------------------
// Pass 2: a_sum[b][k] = sum_t a[b][t][k]; grid B, block 256
// ---------------------------------------------------------------------------
__global__ __launch_bounds__(256) void k_asum(const float* __restrict__ a_in,
                                              float* __restrict__ a_sum)
{
    __shared__ float red[4][64];
    const int tid = threadIdx.x;
    const int b   = blockIdx.x;
    const int k   = tid & 63;
    const int tr  = tid >> 6;
    float s = 0.f;
    for (int t = tr; t < WT; t += 4)
        s += a_in[((size_t)b * WT + t) * WK + k];
    red[tr][k] = s;
    __syncthreads();
    if (tid < 64)
        a_sum[b * WK + tid] = red[0][tid] + red[1][tid] + red[2][tid] + red[3][tid];
}

// ---------------------------------------------------------------------------
// Pass 3: agg[b][k][d-chunk] = sum_t a[b][k][t] * x[b][t][d]
// grid (D/64, B), block 256; WMMA with A = a (M=k), B = x (N=d)
// ---------------------------------------------------------------------------
__global__ __launch_bounds__(256) void k_agg(
    const float* __restrict__ a_in, const float* __restrict__ x,
    float* __restrict__ agg)
{
    __shared__ float aT[64 * PITCH];   // [k][t]  (transposed during staging)
    __shared__ float xT[64 * PITCH];   // [t][d]

    const int tid  = threadIdx.x;
    const int wave = tid >> 5;
    const int lane = tid & 31;
    const int half = lane >> 4;
    const int lm   = lane & 15;
    const int b    = blockIdx.y;
    const int d0   = blockIdx.x * 64;

    const int mt  = wave & 3;                 // k-tile
    const int n0a = ((wave >> 2) * 2) * 16;   // d cols
    const int n0b = n0a + 16;

    v8f acc0 = {}; v8f acc1 = {};

    for (int t0 = 0; t0 < WT; t0 += 64) {
        for (int i = 0; i < 4; ++i) {
            int idx = tid + i * 256;
            int row = idx >> 4;               // t within tile
            int c4  = (idx & 15) << 2;
            f4 xv = *(const f4*)(x + ((size_t)b * WT + t0 + row) * WD + d0 + c4);
            *(f4*)(&xT[row * PITCH + c4]) = xv;
            f4 av = *(const f4*)(a_in + ((size_t)b * WT + t0 + row) * WK + c4);
            aT[(c4 + 0) * PITCH + row] = av.x;
            aT[(c4 + 1) * PITCH + row] = av.y;
            aT[(c4 + 2) * PITCH + row] = av.z;
            aT[(c4 + 3) * PITCH + row] = av.w;
        }
        __syncthreads();
        #pragma unroll
        for (int tt = 0; tt < 64; tt += 4) {
            const int kk = tt + 2 * half;
            v2f af;  af[0]  = aT[(mt * 16 + lm) * PITCH + kk];
                     af[1]  = aT[(mt * 16 + lm) * PITCH + kk + 1];
            v2f bf0; bf0[0] = xT[(kk    ) * PITCH + n0a + lm];
                     bf0[1] = xT[(kk + 1) * PITCH + n0a + lm];
            acc0 = __builtin_amdgcn_wmma_f32_16x16x4_f32(false, af, false, bf0,
                                                         (short)0, acc0, false, false);
            v2f bf1; bf1[0] = xT[(kk    ) * PITCH + n0b + lm];
                     bf1[1] = xT[(kk + 1) * PITCH + n0b + lm];
            acc1 = __builtin_amdgcn_wmma_f32_16x16x4_f32(false, af, false, bf1,
                                                         (short)0, acc1, false, false);
        }
        __syncthreads();
    }
    #pragma unroll
    for (int i = 0; i < 8; ++i) {
        int k = mt * 16 + i + 8 * half;
        agg[((size_t)b * WK + k) * WD + d0 + n0a + lm] = acc0[i];
        agg[((size_t)b * WK + k) * WD + d0 + n0b + lm] = acc1[i];
    }
}

// ---------------------------------------------------------------------------
// Pass 4: per-(b,k) vlad = agg - a_sum*center; intra-cluster L2 normalize.
// grid B*K, block 256 (2 d's per thread). Writes normalized vec to out and
// the cluster's post-normalization squared norm to ssq.
// ---------------------------------------------------------------------------
__global__ __launch_bounds__(256) void k_norm1(
    const float* __restrict__ agg, const float* __restrict__ a_sum,
    const float* __restrict__ centers, float* __restrict__ out,
    float* __restrict__ ssq)
{
    __shared__ float wred[8];
    __shared__ float sinv;
    const int tid = threadIdx.x;
    const int bk  = blockIdx.x;            // b*64 + k
    const int k   = bk & 63;
    const float asum = a_sum[bk];

    float v0 = agg[(size_t)bk * WD + tid      ] - asum * centers[(size_t)k * WD + tid      ];
    float v1 = agg[(size_t)bk * WD + tid + 256] - asum * centers[(size_t)k * WD + tid + 256];
    float p = v0 * v0 + v1 * v1;
    for (int o = 16; o > 0; o >>= 1) p += __shfl_down(p, o, 32);
    if ((tid & 31) == 0) wred[tid >> 5] = p;
    __syncthreads();
    if (tid == 0) {
        float t = 0.f;
        for (int i = 0; i < 8; ++i) t += wred[i];
        float norm  = sqrtf(t);
        float denom = fmaxf(norm, NEPS);
        sinv = 1.0f / denom;
        float r = norm / denom;
        ssq[bk] = r * r;
    }
    __syncthreads();
    const float inv = sinv;
    out[(size_t)bk * WD + tid      ] = v0 * inv;
    out[(size_t)bk * WD + tid + 256] = v1 * inv;
}

// ---------------------------------------------------------------------------
// Pass 5: global L2 over the flattened [K*D] row per b; in-place rescale.
// ---------------------------------------------------------------------------
__global__ __launch_bounds__(256) void k_norm2(float* __restrict__ out,
                                               const float* __restrict__ ssq)
{
    __shared__ float red[64];
    __shared__ float sc;
    const int tid = threadIdx.x;
    const int b   = blockIdx.x;
    if (tid < 64) red[tid] = ssq[b * WK + tid];
    __syncthreads();
    if (tid == 0) {
        float t = 0.f;
        for (int i = 0; i < 64; ++i) t += red[i];
        sc = 1.0f / fmaxf(sqrtf(t), NEPS);
    }
    __syncthreads();
    const float s = sc;
    f4* p = (f4*)(out + (size_t)b * WK * WD);
    for (int i = tid; i < (WK * WD) / 4; i += 256) {
        f4 v = p[i];
        v.x *= s; v.y *= s; v.z *= s; v.w *= s;
        p[i] = v;
    }
}

// ---------------------------------------------------------------------------
extern "C" void kernel_launch(void* const* d_in, const int* in_sizes, int n_in,
                              void* d_out, int out_size, void* d_ws, size_t ws_size,
                              hipStream_t stream)
{
    const float* x       = (const float*)d_in[0];
    const float* centers = (const float*)d_in[1];
    const float* conv_w  = (const float*)d_in[2];
    const float* conv_b  = (const float*)d_in[3];
    float* out = (float*)d_out;

    char* ws = (char*)d_ws;
    size_t off = 0;
    float* a    = (float*)(ws + off); off += (size_t)WB * WT * WK * 4;   // 16 MB
    float* asum = (float*)(ws + off); off += (size_t)WB * WK * 4;        // 8 KB
    float* agg  = (float*)(ws + off); off += (size_t)WB * WK * WD * 4;   // 4 MB
    float* ssq  = (float*)(ws + off); off += (size_t)WB * WK * 4;        // 8 KB

    k_logits_softmax<<<dim3(WT / 64, WB), 256, 0, stream>>>(x, conv_w, conv_b, a);
    k_asum          <<<dim3(WB),          256, 0, stream>>>(a, asum);
    k_agg           <<<dim3(WD / 64, WB), 256, 0, stream>>>(a, x, agg);
    k_norm1         <<<dim3(WB * WK),     256, 0, stream>>>(agg, asum, centers, out, ssq);
    k_norm2         <<<dim3(WB),          256, 0, stream>>>(out, ssq);
}